// RCNPEncoder_65352222376760
// MI455X (gfx1250) — compile-verified
//
#include <hip/hip_runtime.h>
#include <hip/hip_bf16.h>

typedef __attribute__((ext_vector_type(2))) float v2f;
typedef __attribute__((ext_vector_type(8))) float v8f;

#define NC 64
#define NT 64
#define MB 8
#define HD 128           // H == D == 128
#define MT (MB * NT)     // 512 output rows

// ---------------------------------------------------------------------------
// Kernel 1: ReLU double-reduction (VALU-bound core).
//   part[split][m*64+t][h] = sum_{i in split-range, j} relu(p_t + q_i + r_j)
// Block = one (m, t, i-split), 64 threads, each thread owns channels {2*tid, 2*tid+1}.
// r[64] lives in VGPRs (fully unrolled j loop); q_i recomputed per i (3 FMAs,
// amortized over 64 j's); 4 independent accumulator chains for ILP / dual-issue.
// ---------------------------------------------------------------------------
__global__ __launch_bounds__(64) void rcnp_relusum_kernel(
    const float* __restrict__ xc, const float* __restrict__ yc,
    const float* __restrict__ xt, const float* __restrict__ W1,
    const float* __restrict__ b1, float* __restrict__ part, int icnt)
{
  const int m   = blockIdx.x >> 6;
  const int t   = blockIdx.x & 63;
  const int i0  = blockIdx.y * icnt;
  const int tid = threadIdx.x;
  const int h0  = tid << 1;

  __shared__ float s_xc[2 * NC];
  __shared__ float s_yc[NC];
  s_yc[tid]      = yc[m * NC + tid];
  s_xc[tid]      = xc[m * 2 * NC + tid];
  s_xc[NC + tid] = xc[m * 2 * NC + NC + tid];
  __syncthreads();

  // Per-channel weights (2 channels per lane, 8-byte aligned vector loads).
  const float2 wa0 = *(const float2*)&W1[0 * HD + h0];
  const float2 wa1 = *(const float2*)&W1[1 * HD + h0];
  const float2 wb0 = *(const float2*)&W1[2 * HD + h0];
  const float2 wb1 = *(const float2*)&W1[3 * HD + h0];
  const float2 wc  = *(const float2*)&W1[4 * HD + h0];
  const float2 wd  = *(const float2*)&W1[5 * HD + h0];
  const float2 bb  = *(const float2*)&b1[h0];

  const float xt0 = xt[(m * NT + t) * 2 + 0];
  const float xt1 = xt[(m * NT + t) * 2 + 1];

  // p_t = xt·Wa + b1
  const float px = xt0 * wa0.x + xt1 * wa1.x + bb.x;
  const float py = xt0 * wa0.y + xt1 * wa1.y + bb.y;
  // (Wb - Wa) for the q_i term
  const float wbax = wb0.x - wa0.x, wbay = wb0.y - wa0.y;
  const float wbbx = wb1.x - wa1.x, wbby = wb1.y - wa1.y;

  // r_j = yc_j*Wd - xc_j·Wb : register-resident (128 VGPRs).
  float rx[NC], ry[NC];
#pragma unroll
  for (int j = 0; j < NC; ++j) {
    const float xj0 = s_xc[2 * j], xj1 = s_xc[2 * j + 1], yj = s_yc[j];
    rx[j] = yj * wd.x - xj0 * wb0.x - xj1 * wb1.x;
    ry[j] = yj * wd.y - xj0 * wb0.y - xj1 * wb1.y;
  }

  float a0x = 0.f, a1x = 0.f, a0y = 0.f, a1y = 0.f;
  for (int i = i0; i < i0 + icnt; ++i) {
    const float xi0 = s_xc[2 * i], xi1 = s_xc[2 * i + 1], yi = s_yc[i];
    const float cx = px + xi0 * wbax + xi1 * wbbx + yi * wc.x;
    const float cy = py + xi0 * wbay + xi1 * wbby + yi * wc.y;
#pragma unroll
    for (int j = 0; j < NC; j += 2) {
      a0x += fmaxf(cx + rx[j], 0.f);
      a0y += fmaxf(cy + ry[j], 0.f);
      a1x += fmaxf(cx + rx[j + 1], 0.f);
      a1y += fmaxf(cy + ry[j + 1], 0.f);
    }
  }

  const int row = m * NT + t;
  float2 res;
  res.x = a0x + a1x;
  res.y = a0y + a1y;
  *(float2*)&part[(size_t)blockIdx.y * (MT * HD) + row * HD + h0] = res;
}

// ---------------------------------------------------------------------------
// Kernel 2: out(512x128) = hsum(512x128) @ W2(128x128) + nc^2 * b2
// Full-fp32 GEMM on V_WMMA_F32_16X16X4_F32. One wave per 16x16 tile,
// 32 k-steps. A layout: lane-lo K={0,1}, lane-hi K={2,3}; B mirrored;
// C/D: VGPR g -> rows g / g+8 across lane halves, N = lane&15.
// NSPLIT is a template parameter so the partial-fold is straight-line code
// (no per-k-step SALU/branch churn in the unrolled loop).
// ---------------------------------------------------------------------------
template <int NSPLIT>
__global__ __launch_bounds__(128) void rcnp_out_gemm_kernel(
    const float* __restrict__ Hs, const float* __restrict__ W2,
    const float* __restrict__ b2, float* __restrict__ out)
{
  const int wave = threadIdx.x >> 5;
  const int lane = threadIdx.x & 31;
  const int tile = blockIdx.x * 4 + wave;  // 256 tiles total (32 M-tiles x 8 N-tiles)
  const int tM = tile >> 3;
  const int tN = tile & 7;
  const int half = lane >> 4;
  const int l16  = lane & 15;

  const int arow = tM * 16 + l16;  // A row for this lane
  const int col  = tN * 16 + l16;  // B / D column for this lane

  const float* Aptr  = Hs + arow * HD + 2 * half;                 // partial 0
  const float* Aptr2 = Aptr + MT * HD;                            // partial 1
  const float* Bptr  = W2 + (2 * half) * HD + col;

  v8f acc = {};
#pragma unroll
  for (int k = 0; k < HD / 4; ++k) {
    float2 av = *(const float2*)(Aptr + 4 * k);
    if (NSPLIT == 2) {  // compile-time: folds to an unconditional b64 load + pk_add
      const float2 av2 = *(const float2*)(Aptr2 + 4 * k);
      av.x += av2.x;
      av.y += av2.y;
    }
    v2f a;
    a.x = av.x;                     // A[arow][4k + 2*half + 0]
    a.y = av.y;                     // A[arow][4k + 2*half + 1]
    v2f b;
    b.x = Bptr[(4 * k) * HD];       // B[4k + 2*half + 0][col]
    b.y = Bptr[(4 * k) * HD + HD];  // B[4k + 2*half + 1][col]
    acc = __builtin_amdgcn_wmma_f32_16x16x4_f32(
        /*neg_a=*/false, a, /*neg_b=*/false, b,
        /*c_mod=*/(short)0, acc, /*reuse_a=*/false, /*reuse_b=*/false);
  }

  const float bias = (float)(NC * NC) * b2[col];
#pragma unroll
  for (int g = 0; g < 8; ++g)
    out[(tM * 16 + g + 8 * half) * HD + col] = acc[g] + bias;
}

// ---------------------------------------------------------------------------
extern "C" void kernel_launch(void* const* d_in, const int* in_sizes, int n_in,
                              void* d_out, int out_size, void* d_ws, size_t ws_size,
                              hipStream_t stream) {
  (void)in_sizes; (void)n_in; (void)out_size;
  const float* xc = (const float*)d_in[0];
  const float* yc = (const float*)d_in[1];
  const float* xt = (const float*)d_in[2];
  const float* W1 = (const float*)d_in[3];
  const float* b1 = (const float*)d_in[4];
  const float* W2 = (const float*)d_in[5];
  const float* b2 = (const float*)d_in[6];
  float* out = (float*)d_out;
  float* ws  = (float*)d_ws;

  const size_t partBytes = (size_t)MT * HD * sizeof(float);  // 256 KB per partial
  const int nsplit = (ws_size >= 2 * partBytes) ? 2 : 1;     // constant across calls -> deterministic
  const int icnt = NC / nsplit;

  dim3 grid1(MB * NT, nsplit);  // 512 (m,t) blocks x i-splits
  rcnp_relusum_kernel<<<grid1, 64, 0, stream>>>(xc, yc, xt, W1, b1, ws, icnt);
  if (nsplit == 2)
    rcnp_out_gemm_kernel<2><<<64, 128, 0, stream>>>(ws, W2, b2, out);
  else
    rcnp_out_gemm_kernel<1><<<64, 128, 0, stream>>>(ws, W2, b2, out);
}